// DQGN_37847251812370
// MI455X (gfx1250) — compile-verified
//
#include <hip/hip_runtime.h>
#include <hip/hip_bf16.h>

// ---------------------------------------------------------------------------
// GCN on MI455X (gfx1250): fp32 WMMA (v_wmma_f32_16x16x4_f32) for dense GEMMs,
// global_atomic_add_f32 for the edge scatter (the true bottleneck: ~3.7 GB of
// edge traffic/дcall vs only 4.5 GFLOP of GEMM).
// ---------------------------------------------------------------------------

typedef __attribute__((ext_vector_type(2))) float v2f;
typedef __attribute__((ext_vector_type(8))) float v8f;

#define NNODES 50000
#define NEDGES 800000
#define CIN 128
#define CH 64
#define NLAYERS 9
#define NGRAPH 10
#define GNODES 5000

// ----------------------------- utility fills -------------------------------
__global__ void fill_kernel(float* __restrict__ p, float v, int n) {
  int t = blockIdx.x * blockDim.x + threadIdx.x;
  if (t < n) p[t] = v;
}

// ------------------------- degree / normalization --------------------------
__global__ void degree_kernel(const int* __restrict__ dst, float* __restrict__ deg, int e) {
  int t = blockIdx.x * blockDim.x + threadIdx.x;
  if (t < e) atomicAdd(&deg[dst[t]], 1.0f);
}

__global__ void rsqrt_kernel(float* __restrict__ d, int n) {
  int t = blockIdx.x * blockDim.x + threadIdx.x;
  if (t < n) d[t] = rsqrtf(d[t]);
}

__global__ void enorm_kernel(const int* __restrict__ src, const int* __restrict__ dst,
                             const float* __restrict__ dinv, float* __restrict__ enorm, int e) {
  int t = blockIdx.x * blockDim.x + threadIdx.x;
  if (t < e) enorm[t] = dinv[src[t]] * dinv[dst[t]];
}

// ------------------------------ WMMA GEMM ----------------------------------
// Out[N,64] = A[N,K] @ W[K,64], fp32 via V_WMMA_F32_16X16X4_F32.
// 8 waves / block, each wave owns a 16-row x 64-col strip (4 accumulators).
// A fragment (16x4): lanes 0-15 hold (M=lane, K=k0,k0+1); lanes 16-31 hold K=k0+2,k0+3.
// B fragment (4x16): same pattern with N = lane&15.
// C/D: VGPR i -> M = i + 8*(lane>=16), N = lane&15.
__global__ __launch_bounds__(256) void gemm_wmma_kernel(
    const float* __restrict__ A, const float* __restrict__ W,
    float* __restrict__ Out, int nrows, int K) {
  const int lane = threadIdx.x & 31;
  const int wave = threadIdx.x >> 5;
  const int rowTile = blockIdx.x * 8 + wave;
  const int row0 = rowTile * 16;
  if (row0 >= nrows) return;

  const int half = lane >> 4;   // 0: K-pair (k,k+1), 1: K-pair (k+2,k+3)
  const int l16  = lane & 15;
  const int arow = row0 + l16;

  v8f acc0 = {}, acc1 = {}, acc2 = {}, acc3 = {};

  for (int k = 0; k < K; k += 4) {
    const int ka = k + half * 2;
    v2f a;
    a.x = A[arow * K + ka];
    a.y = A[arow * K + ka + 1];

    v2f b0, b1, b2, b3;
    const float* wr0 = W + ka * CH;         // row ka
    const float* wr1 = W + (ka + 1) * CH;   // row ka+1
    b0.x = wr0[l16];      b0.y = wr1[l16];
    b1.x = wr0[16 + l16]; b1.y = wr1[16 + l16];
    b2.x = wr0[32 + l16]; b2.y = wr1[32 + l16];
    b3.x = wr0[48 + l16]; b3.y = wr1[48 + l16];

    acc0 = __builtin_amdgcn_wmma_f32_16x16x4_f32(false, a, false, b0, (short)0, acc0, false, false);
    acc1 = __builtin_amdgcn_wmma_f32_16x16x4_f32(false, a, false, b1, (short)0, acc1, false, false);
    acc2 = __builtin_amdgcn_wmma_f32_16x16x4_f32(false, a, false, b2, (short)0, acc2, false, false);
    acc3 = __builtin_amdgcn_wmma_f32_16x16x4_f32(false, a, false, b3, (short)0, acc3, false, false);
  }

#pragma unroll
  for (int i = 0; i < 8; ++i) {
    const int m = i + half * 8;
    float* orow = Out + (size_t)(row0 + m) * CH + l16;
    orow[0]  = acc0[i];
    orow[16] = acc1[i];
    orow[32] = acc2[i];
    orow[48] = acc3[i];
  }
}

// ------------------------------ edge scatter -------------------------------
// agg[dst] += hw[src] * enorm[e]; one wave per edge, float2 per lane (64 ch).
__global__ __launch_bounds__(256) void scatter_kernel(
    const float* __restrict__ hw, const float* __restrict__ enorm,
    const int* __restrict__ src, const int* __restrict__ dst,
    float* __restrict__ agg, int e) {
  const int t = blockIdx.x * blockDim.x + threadIdx.x;
  const int eid = t >> 5;
  if (eid >= e) return;
  const int c = (t & 31) * 2;
  const float w = enorm[eid];
  const int s = src[eid];
  const int d = dst[eid];
  const float2 v = *(const float2*)(hw + (size_t)s * CH + c);
  float* ap = agg + (size_t)d * CH + c;
  atomicAdd(ap,     v.x * w);
  atomicAdd(ap + 1, v.y * w);
}

// ---------------------- combine: relu(agg + hw*snorm + b) ------------------
__global__ void combine_kernel(const float* __restrict__ agg, const float* __restrict__ hw,
                               const float* __restrict__ dinv, const float* __restrict__ bias,
                               float* __restrict__ hout, int n) {
  const int t = blockIdx.x * blockDim.x + threadIdx.x;
  if (t >= n * CH) return;
  const int node = t >> 6;
  const int c = t & 63;
  const float di = dinv[node];
  const float v = agg[t] + hw[t] * (di * di) + bias[c];
  hout[t] = fmaxf(v, 0.0f);
}

// ------------------------------- pooling -----------------------------------
// Per-graph global add pool; LDS partials, one atomic per channel per block.
__global__ __launch_bounds__(256) void pool_kernel(const float* __restrict__ h2,
                                                   float* __restrict__ pool) {
  const int g = blockIdx.y;
  const int chunk = blockIdx.x;           // 4 chunks of 1250 nodes
  const int c = threadIdx.x & 63;
  const int r = threadIdx.x >> 6;         // 0..3
  float s = 0.0f;
  const int base = g * GNODES + chunk * (GNODES / 4);
  for (int i = r; i < GNODES / 4; i += 4)
    s += h2[(size_t)(base + i) * CH + c];
  __shared__ float sm[4][CH];
  sm[r][c] = s;
  __syncthreads();
  if (r == 0)
    atomicAdd(&pool[g * CH + c], sm[0][c] + sm[1][c] + sm[2][c] + sm[3][c]);
}

// rep = pool @ Wp  (10x64 @ 64x64 — tiny)
__global__ void repmm_kernel(const float* __restrict__ pool, const float* __restrict__ Wp,
                             float* __restrict__ rep) {
  const int t = blockIdx.x * blockDim.x + threadIdx.x;
  if (t >= NGRAPH * CH) return;
  const int g = t >> 6, c = t & 63;
  float s = 0.0f;
  for (int k = 0; k < CH; ++k) s += pool[g * CH + k] * Wp[k * CH + c];
  rep[t] = s;
}

// out[n] = tanh( relu(h2[n]).Wa[0:64] + relu(rep[g]).Wa[64:128] ); wave per node.
__global__ __launch_bounds__(256) void final_kernel(const float* __restrict__ h2,
                                                    const float* __restrict__ rep,
                                                    const float* __restrict__ Wa,
                                                    float* __restrict__ out, int n) {
  const int t = blockIdx.x * blockDim.x + threadIdx.x;
  const int node = t >> 5;
  if (node >= n) return;
  const int lane = t & 31;
  const int g = node / GNODES;
  const float* hr = h2 + (size_t)node * CH;
  const float* rr = rep + (size_t)g * CH;
  float s = fmaxf(hr[lane], 0.0f)      * Wa[lane]
          + fmaxf(hr[lane + 32], 0.0f) * Wa[lane + 32]
          + fmaxf(rr[lane], 0.0f)      * Wa[64 + lane]
          + fmaxf(rr[lane + 32], 0.0f) * Wa[96 + lane];
#pragma unroll
  for (int off = 16; off > 0; off >>= 1)
    s += __shfl_xor(s, off, 32);
  if (lane == 0) out[node] = tanhf(s);
}

// ------------------------------- host side ---------------------------------
extern "C" void kernel_launch(void* const* d_in, const int* in_sizes, int n_in,
                              void* d_out, int out_size, void* d_ws, size_t ws_size,
                              hipStream_t stream) {
  const float* x   = (const float*)d_in[0];     // [N, 128]
  const int*   ei  = (const int*)d_in[1];       // [2, E]
  const float* W0  = (const float*)d_in[2];     // [128, 64]
  const float* b0  = (const float*)d_in[3];     // [64]
  const float* Ws  = (const float*)d_in[4];     // [8, 64, 64]
  const float* bs  = (const float*)d_in[5];     // [8, 64]
  const float* Wn  = (const float*)d_in[6];     // [64, 64]
  const float* Wp  = (const float*)d_in[7];     // [64, 64]
  const float* Wa  = (const float*)d_in[8];     // [128, 1]
  float* out = (float*)d_out;                   // [N, 1]

  const int* srcp = ei;
  const int* dstp = ei + NEDGES;

  // workspace layout (floats)
  float* w     = (float*)d_ws;
  float* dinv  = w;                              // NNODES      (deg -> rsqrt)
  float* enorm = dinv + 51200;                   // NEDGES
  float* hA    = enorm + NEDGES;                 // N*64
  float* hB    = hA + (size_t)NNODES * CH;       // N*64
  float* hw    = hB + (size_t)NNODES * CH;       // N*64
  float* agg   = hw + (size_t)NNODES * CH;       // N*64
  float* pool  = agg + (size_t)NNODES * CH;      // 10*64
  float* rep   = pool + NGRAPH * CH;             // 10*64

  const int NC = NNODES * CH;

  // --- degree + symmetric norm (self loops: deg starts at 1.0) ---
  fill_kernel<<<(NNODES + 255) / 256, 256, 0, stream>>>(dinv, 1.0f, NNODES);
  degree_kernel<<<(NEDGES + 255) / 256, 256, 0, stream>>>(dstp, dinv, NEDGES);
  rsqrt_kernel<<<(NNODES + 255) / 256, 256, 0, stream>>>(dinv, NNODES);
  enorm_kernel<<<(NEDGES + 255) / 256, 256, 0, stream>>>(srcp, dstp, dinv, enorm, NEDGES);

  // --- 9 GCN layers ---
  const float* hin = x;
  float* cur = hA;
  for (int i = 0; i < NLAYERS; ++i) {
    const int K = (i == 0) ? CIN : CH;
    const float* Wl = (i == 0) ? W0 : Ws + (size_t)(i - 1) * CH * CH;
    const float* bl = (i == 0) ? b0 : bs + (size_t)(i - 1) * CH;
    float* hout = (i & 1) ? hB : hA;

    const int rowTiles = NNODES / 16;   // 3125
    gemm_wmma_kernel<<<(rowTiles + 7) / 8, 256, 0, stream>>>(hin, Wl, hw, NNODES, K);
    fill_kernel<<<(NC + 255) / 256, 256, 0, stream>>>(agg, 0.0f, NC);
    scatter_kernel<<<((NEDGES * 32) + 255) / 256, 256, 0, stream>>>(hw, enorm, srcp, dstp, agg, NEDGES);
    combine_kernel<<<(NC + 255) / 256, 256, 0, stream>>>(agg, hw, dinv, bl, hout, NNODES);

    hin = hout;
    cur = hout;
  }

  // --- node transform: h2 = h @ Wn (reuse hw buffer) ---
  {
    const int rowTiles = NNODES / 16;
    gemm_wmma_kernel<<<(rowTiles + 7) / 8, 256, 0, stream>>>(cur, Wn, hw, NNODES, CH);
  }
  float* h2 = hw;

  // --- global add pool per graph, rep = pool @ Wp ---
  fill_kernel<<<(NGRAPH * CH + 255) / 256, 256, 0, stream>>>(pool, 0.0f, NGRAPH * CH);
  pool_kernel<<<dim3(4, NGRAPH), 256, 0, stream>>>(h2, pool);
  repmm_kernel<<<(NGRAPH * CH + 255) / 256, 256, 0, stream>>>(pool, Wp, rep);

  // --- final: tanh(relu(cat(h2, rep[batch])) @ Wa) ---
  final_kernel<<<((NNODES * 32) + 255) / 256, 256, 0, stream>>>(h2, rep, Wa, out, NNODES);
}